// ConcatHeadModule_62852551409784
// MI455X (gfx1250) — compile-verified
//
#include <hip/hip_runtime.h>
#include <math.h>

// ---------------------------------------------------------------------------
// Problem constants (from the reference): T=768, LDIMS=256, HID=HID2=512.
// ---------------------------------------------------------------------------
#define T_DIM   768
#define HIDN    512
#define KDIM    512          // 2*LDIMS — inner dim of the first-layer GEMMs

typedef float v2f __attribute__((ext_vector_type(2)));
typedef float v8f __attribute__((ext_vector_type(8)));

// Hardware tanh (V_TANH_F32, TRANS pipe) when the builtin exists; libm otherwise.
#if defined(__has_builtin)
#if __has_builtin(__builtin_amdgcn_tanhf)
#define TANH(x) __builtin_amdgcn_tanhf(x)
#endif
#endif
#ifndef TANH
#define TANH(x) tanhf(x)
#endif

// ---------------------------------------------------------------------------
// Fp32 WMMA GEMM:  C[M,N] = act( A[M,K] @ B[K,N] + bias[N] )
//   - one wave computes one 16x16 output tile via V_WMMA_F32_16X16X4_F32
//     (full fp32 precision, matching the fp32 reference)
//   - 4 waves (128 threads) per block; all wmma-adjacent branches wave-uniform
// Fragment layouts per CDNA5 ISA 7.12.2:
//   A 16x4 : lanes 0-15 hold (M=lane, K=k..k+1), lanes 16-31 hold (M, K=k+2..k+3)
//   B 4x16 : VGPR v, lane L -> B[k + v + 2*(L>>4)][col0 + (L&15)]
//   C 16x16: VGPR v, lane L -> C[row0 + v + 8*(L>>4)][col0 + (L&15)]
// ---------------------------------------------------------------------------
__global__ __launch_bounds__(128)
void wmma_gemm_bias_act(const float* __restrict__ A,
                        const float* __restrict__ B,
                        const float* __restrict__ bias,   // length N, or null
                        float* __restrict__ C,
                        int M, int N, int K, int applyTanh)
{
    const int lane   = threadIdx.x & 31;
    const int wave   = threadIdx.x >> 5;
    const int tile   = blockIdx.x * 4 + wave;
    const int ntn    = N >> 4;                 // tiles along N
    const int trow   = tile / ntn;
    const int tcol   = tile - trow * ntn;
    if (trow >= (M >> 4)) return;              // wave-uniform: EXEC stays all-1s

    const int row0   = trow << 4;
    const int col0   = tcol << 4;
    const int lhalf  = lane >> 4;              // 0 or 1
    const int l16    = lane & 15;

    // Strength-reduced pointers: advance by constants each k-step.
    const float* aPtr = A + (size_t)(row0 + l16) * K + 2 * lhalf;
    const float* b0   = B + (size_t)(2 * lhalf) * N + col0 + l16;
    const float* b1   = b0 + N;
    const size_t bInc = 4 * (size_t)N;

    v8f acc = {};
    for (int k = 0; k < K; k += 4) {
        v2f a = *(const v2f*)aPtr;             // (K even) -> 8B aligned
        v2f b;
        b.x = *b0;
        b.y = *b1;
        aPtr += 4;
        b0   += bInc;
        b1   += bInc;
        // 8 args: (neg_a, A, neg_b, B, c_mod, C, reuse_a, reuse_b)
        acc = __builtin_amdgcn_wmma_f32_16x16x4_f32(
                  false, a, false, b, (short)0, acc, false, false);
    }

    const float bv = bias ? bias[col0 + l16] : 0.0f;
    float* Cout = C + col0 + l16;
    #pragma unroll
    for (int v = 0; v < 8; ++v) {
        const int m = row0 + v + 8 * lhalf;
        float r = acc[v] + bv;
        if (applyTanh) r = TANH(r);
        Cout[(size_t)m * N] = r;
    }
}

// ---------------------------------------------------------------------------
// Pairwise stage: out[i,j] = outBias + sum_h w[h] * tanh(AH[i,h] + AMb[j,h])
// 302M tanh total -> V_TANH_F32 (TRANS) throughput bound. 64(i) x 32(j) tile
// per 256-thread block; h staged through LDS in chunks of 32, stored
// TRANSPOSED (h-major) so each thread's inner step is:
//   1x ds_load_b128 (4 a's) + 1x ds_load_b64 (2 b's) + 1 broadcast (w)
//   -> 8 independent tanh+fma chains per h (good DS:TRANS ratio + ILP).
// Padding: PA=68, PB=36 keep 16B/8B alignment and <=2-way store conflicts.
// ---------------------------------------------------------------------------
#define TI 64
#define TJ 32
#define HC 32
#define PA (TI + 4)
#define PB (TJ + 4)

__global__ __launch_bounds__(256)
void pairwise_tanh_reduce(const float* __restrict__ AH,
                          const float* __restrict__ AM,
                          const float* __restrict__ w,       // outLayer[:,0]
                          const float* __restrict__ outBias, // single element
                          float* __restrict__ out)
{
    __shared__ float sAt[HC][PA];   // sAt[h][i]  (transposed)
    __shared__ float sBt[HC][PB];   // sBt[h][j]  (transposed)
    __shared__ float sw[HC];

    const int tid = threadIdx.x;
    const int i0  = blockIdx.x * TI;
    const int j0  = blockIdx.y * TJ;
    const int ig4 = (tid >> 4) << 2;   // 0,4,...,60 : first of 4 rows
    const int jg2 = (tid & 15) << 1;   // 0,2,...,30 : first of 2 cols

    float a0x = 0.f, a0y = 0.f, a1x = 0.f, a1y = 0.f;
    float a2x = 0.f, a2y = 0.f, a3x = 0.f, a3y = 0.f;

    for (int h0 = 0; h0 < HIDN; h0 += HC) {
        __syncthreads();
        #pragma unroll
        for (int idx = tid; idx < TI * HC; idx += 256) {
            const int r = idx >> 5;        // HC == 32
            const int c = idx & 31;
            sAt[c][r] = AH[(size_t)(i0 + r) * HIDN + h0 + c];
        }
        #pragma unroll
        for (int idx = tid; idx < TJ * HC; idx += 256) {
            const int r = idx >> 5;
            const int c = idx & 31;
            sBt[c][r] = AM[(size_t)(j0 + r) * HIDN + h0 + c];
        }
        if (tid < HC) sw[tid] = w[h0 + tid];
        __syncthreads();

        #pragma unroll 4
        for (int h = 0; h < HC; ++h) {
            const float4 a = *(const float4*)&sAt[h][ig4];  // 16B aligned
            const float2 b = *(const float2*)&sBt[h][jg2];  // 8B aligned
            const float wv = sw[h];
            a0x += wv * TANH(a.x + b.x);
            a0y += wv * TANH(a.x + b.y);
            a1x += wv * TANH(a.y + b.x);
            a1y += wv * TANH(a.y + b.y);
            a2x += wv * TANH(a.z + b.x);
            a2y += wv * TANH(a.z + b.y);
            a3x += wv * TANH(a.w + b.x);
            a3y += wv * TANH(a.w + b.y);
        }
    }

    const float ob = outBias[0];
    float* o = out + (size_t)(i0 + ig4) * T_DIM + j0 + jg2;
    *(float2*)(o)             = make_float2(a0x + ob, a0y + ob);
    *(float2*)(o + T_DIM)     = make_float2(a1x + ob, a1y + ob);
    *(float2*)(o + 2 * T_DIM) = make_float2(a2x + ob, a2y + ob);
    *(float2*)(o + 3 * T_DIM) = make_float2(a3x + ob, a3y + ob);
}

// ---------------------------------------------------------------------------
// Host-side launch. Inputs (all float32, setup_inputs order):
//  0 x            [768, 2, 256]   (contiguous == xc [768,512])
//  1 hidLayerFOH  [512, 512]
//  2 hidLayerFOM  [512, 512]
//  3 catBias      [1024]
//  4 hid2Layer    [1024, 512]
//  5 hid2Bias     [512]
//  6 outLayer     [512, 1]
//  7 outBias      [1]
// Output: scores [768, 768] float32.
// Workspace: 4 buffers of 768*512 floats = 6 MB.
// ---------------------------------------------------------------------------
extern "C" void kernel_launch(void* const* d_in, const int* in_sizes, int n_in,
                              void* d_out, int out_size, void* d_ws, size_t ws_size,
                              hipStream_t stream)
{
    (void)in_sizes; (void)n_in; (void)out_size; (void)ws_size;

    const float* x       = (const float*)d_in[0];
    const float* Wfoh    = (const float*)d_in[1];
    const float* Wfom    = (const float*)d_in[2];
    const float* catBias = (const float*)d_in[3];
    const float* W2      = (const float*)d_in[4];
    const float* b2      = (const float*)d_in[5];
    const float* wout    = (const float*)d_in[6];
    const float* outBias = (const float*)d_in[7];
    float*       out     = (float*)d_out;

    float* ws   = (float*)d_ws;
    const size_t MAT = (size_t)T_DIM * HIDN;   // 768*512
    float* actH = ws;
    float* actM = ws + MAT;
    float* AH   = ws + 2 * MAT;
    float* AMb  = ws + 3 * MAT;

    // 768/16 * 512/16 = 1536 tiles; 4 waves per block -> 384 blocks
    const int gemmBlocks = ((T_DIM / 16) * (HIDN / 16)) / 4;

    // Stage 1: actH = tanh(xc @ Wfoh + catBias[:512]); actM = tanh(xc @ Wfom + catBias[512:])
    wmma_gemm_bias_act<<<gemmBlocks, 128, 0, stream>>>(
        x, Wfoh, catBias,        actH, T_DIM, HIDN, KDIM, 1);
    wmma_gemm_bias_act<<<gemmBlocks, 128, 0, stream>>>(
        x, Wfom, catBias + HIDN, actM, T_DIM, HIDN, KDIM, 1);

    // Stage 2: AH = actH @ W2[:512];  AMb = actM @ W2[512:] + hid2Bias
    wmma_gemm_bias_act<<<gemmBlocks, 128, 0, stream>>>(
        actH, W2,                       nullptr, AH,  T_DIM, HIDN, HIDN, 0);
    wmma_gemm_bias_act<<<gemmBlocks, 128, 0, stream>>>(
        actM, W2 + (size_t)HIDN * HIDN, b2,      AMb, T_DIM, HIDN, HIDN, 0);

    // Stage 3: out[i,j] = outBias + sum_h wout[h]*tanh(AH[i,h] + AMb[j,h])
    dim3 pg(T_DIM / TI, T_DIM / TJ);
    pairwise_tanh_reduce<<<pg, 256, 0, stream>>>(AH, AMb, wout, outBias, out);
}